// MultiScaleRRWPFiltration_65867618452257
// MI455X (gfx1250) — compile-verified
//
#include <hip/hip_runtime.h>
#include <hip/hip_bf16.h>

// Problem constants (match the reference)
#define N_     1024
#define E_     32768
#define S_     4
#define STEPS_ 31
#define FEAT_  (S_ * STEPS_)   // 124

typedef __attribute__((ext_vector_type(2))) float v2f;
typedef __attribute__((ext_vector_type(8))) float v8f;

// ---------------------------------------------------------------------------
// Kernel 1: build P (S x N x N), row-stochastic Gaussian affinity, zero diag.
// One block per (sigma, row). 256 threads, each handles 4 columns.
// ---------------------------------------------------------------------------
__global__ void __launch_bounds__(256)
build_p_kernel(const float* __restrict__ coords,
               const float* __restrict__ log_sigmas,
               float* __restrict__ P)
{
    const int b = blockIdx.x;          // 0 .. S*N-1
    const int s = b >> 10;             // / N
    const int i = b & (N_ - 1);        // % N

    // sigmas = sort(exp(log_sigmas)) — S=4, tiny insertion sort in registers
    float sig[S_];
#pragma unroll
    for (int k = 0; k < S_; ++k) sig[k] = expf(log_sigmas[k]);
#pragma unroll
    for (int a = 1; a < S_; ++a) {
        float x = sig[a];
        int c = a - 1;
        while (c >= 0 && sig[c] > x) { sig[c + 1] = sig[c]; --c; }
        sig[c + 1] = x;
    }
    const float inv = -1.0f / (2.0f * sig[s] * sig[s]);

    const float cx = coords[3 * i + 0];
    const float cy = coords[3 * i + 1];
    const float cz = coords[3 * i + 2];

    __shared__ float red[256];
    float wv[4];
    float partial = 0.0f;
#pragma unroll
    for (int r = 0; r < 4; ++r) {
        const int j = threadIdx.x + (r << 8);
        const float dx = cx - coords[3 * j + 0];
        const float dy = cy - coords[3 * j + 1];
        const float dz = cz - coords[3 * j + 2];
        const float d2 = dx * dx + dy * dy + dz * dz;
        const float w = (j == i) ? 0.0f : expf(d2 * inv);
        wv[r] = w;
        partial += w;
    }
    red[threadIdx.x] = partial;
    __syncthreads();
#pragma unroll
    for (int off = 128; off > 0; off >>= 1) {
        if ((int)threadIdx.x < off) red[threadIdx.x] += red[threadIdx.x + off];
        __syncthreads();
    }
    const float invrs = 1.0f / fmaxf(red[0], 1e-12f);

    float* __restrict__ row = P + ((size_t)s * N_ + i) * N_;
#pragma unroll
    for (int r = 0; r < 4; ++r) {
        const int j = threadIdx.x + (r << 8);
        row[j] = wv[r] * invrs;
    }
}

// ---------------------------------------------------------------------------
// Kernel 2: batched fp32 GEMM step: D[s] = A[s] @ B[s], s = blockIdx.z.
// 128x128 block tile, 256 threads = 8 waves (2 across M x 4 across N),
// wave tile 64x32 = 4x2 v_wmma_f32_16x16x4_f32 fragments.
// Global->LDS via CDNA5 async-to-LDS (ASYNCcnt), double-buffered so the
// next K-slab's fetch overlaps the current slab's WMMAs.
//
// Fragment layouts (ISA 7.12.2, wave32):
//   A 16x4:  lane l: M = l&15, half = l>>4 ; vgpr v holds A[M][2*half + v]
//   B 4x16:  lane l: Nc = l&15, half = l>>4; vgpr v holds B[2*half + v][Nc]
//   C 16x16: lane l: Nc = l&15, half = l>>4; vgpr v holds C[v + 8*half][Nc]
// ---------------------------------------------------------------------------
#define TM 128
#define TN 128
#define TK 32
#define APITCH (TK + 4)   // 36 floats: 144B rows (16B aligned), frag reads conflict-free
#define BPITCH (TN + 8)   // 136 floats: 544B rows (16B aligned), halves on disjoint banks

__global__ void __launch_bounds__(256)
gemm_step_kernel(const float* __restrict__ A,   // S x N x N (P_cur)
                 const float* __restrict__ B,   // S x N x N (P)
                 float* __restrict__ D)         // S x N x N (P_next)
{
    const int s     = blockIdx.z;
    const size_t mo = (size_t)s * N_ * N_;
    const int row0  = blockIdx.y * TM;
    const int col0  = blockIdx.x * TN;

    __shared__ __align__(16) float As[2][TM][APITCH];
    __shared__ __align__(16) float Bs[2][TK][BPITCH];

    const int tid  = threadIdx.x;
    const int lane = tid & 31;
    const int wave = tid >> 5;
    const int wm   = wave & 1;         // M half of block tile (64 rows)
    const int wn   = wave >> 1;        // N quarter of block tile (32 cols)
    const int l15  = lane & 15;
    const int half = lane >> 4;

    v8f acc[4][2] = {};

    // Async prefetch of one K-slab (A: 128x32, B: 32x128) into LDS buffer `buf`.
    // Each slab = 1024 16B chunks; 256 threads issue 4 async b128 loads each.
    auto prefetch = [&](int buf, int kb) {
        const uint32_t aBase = (uint32_t)(uintptr_t)&As[buf][0][0];
        const uint32_t bBase = (uint32_t)(uintptr_t)&Bs[buf][0][0];
#pragma unroll
        for (int r = 0; r < 4; ++r) {
            const int c = tid + (r << 8);

            const int ar = c >> 3;              // A row 0..127
            const int ac = (c & 7) << 2;        // A col 0,4,..,28
            const float* ga = A + mo + (size_t)(row0 + ar) * N_ + kb + ac;
            const uint32_t la = aBase + (uint32_t)(ar * APITCH + ac) * 4u;
            asm volatile("global_load_async_to_lds_b128 %0, %1, off"
                         :: "v"(la), "v"(ga) : "memory");

            const int br = c >> 5;              // B row 0..31
            const int bc = (c & 31) << 2;       // B col 0,4,..,124
            const float* gb = B + mo + (size_t)(kb + br) * N_ + col0 + bc;
            const uint32_t lb = bBase + (uint32_t)(br * BPITCH + bc) * 4u;
            asm volatile("global_load_async_to_lds_b128 %0, %1, off"
                         :: "v"(lb), "v"(gb) : "memory");
        }
    };

    prefetch(0, 0);
    asm volatile("s_wait_asynccnt 0x0" ::: "memory");
    __syncthreads();

    for (int kb = 0; kb < N_; kb += TK) {
        const int cur = (kb >> 5) & 1;
        if (kb + TK < N_) prefetch(cur ^ 1, kb + TK);   // overlap with WMMAs below

#pragma unroll
        for (int k = 0; k < TK; k += 4) {
            v2f aF[4], bF[2];
#pragma unroll
            for (int mi = 0; mi < 4; ++mi) {
                const float* ap = &As[cur][wm * 64 + mi * 16 + l15][k + 2 * half];
                aF[mi].x = ap[0];
                aF[mi].y = ap[1];
            }
#pragma unroll
            for (int ni = 0; ni < 2; ++ni) {
                const int nc = wn * 32 + ni * 16 + l15;
                bF[ni].x = Bs[cur][k + 2 * half + 0][nc];
                bF[ni].y = Bs[cur][k + 2 * half + 1][nc];
            }
#pragma unroll
            for (int mi = 0; mi < 4; ++mi)
#pragma unroll
                for (int ni = 0; ni < 2; ++ni)
                    acc[mi][ni] = __builtin_amdgcn_wmma_f32_16x16x4_f32(
                        false, aF[mi], false, bF[ni],
                        (short)0, acc[mi][ni], false, false);
        }

        // Own async prefetch done -> barrier makes whole next slab visible,
        // and licenses overwrite of the buffer we just read next iteration.
        asm volatile("s_wait_asynccnt 0x0" ::: "memory");
        __syncthreads();
    }

    // Epilogue: scatter 4x2 fragments to D.
#pragma unroll
    for (int mi = 0; mi < 4; ++mi)
#pragma unroll
        for (int ni = 0; ni < 2; ++ni) {
            const int col = col0 + wn * 32 + ni * 16 + l15;
#pragma unroll
            for (int v = 0; v < 8; ++v) {
                const int row = row0 + wm * 64 + mi * 16 + v + 8 * half;
                D[mo + (size_t)row * N_ + col] = acc[mi][ni][v];
            }
        }
}

// ---------------------------------------------------------------------------
// Kernel 3: extract diag + edge features for power step t (0-based).
// node_features[i, s*STEPS + t] = M[s][i][i]
// edge_features[e, s*STEPS + t] = M[s][ei[e]][ej[e]]
// ---------------------------------------------------------------------------
__global__ void __launch_bounds__(256)
extract_kernel(const float* __restrict__ M,
               const int* __restrict__ eidx,   // (2, E) flat: ei then ej
               float* __restrict__ out, int t)
{
    const int SN    = S_ * N_;            // 4096
    const int total = SN + S_ * E_;       // 135168
    const int idx = blockIdx.x * blockDim.x + threadIdx.x;
    if (idx >= total) return;

    if (idx < SN) {
        const int s = idx >> 10;
        const int i = idx & (N_ - 1);
        const float v = M[((size_t)s * N_ + i) * N_ + i];
        out[(size_t)i * FEAT_ + s * STEPS_ + t] = v;
    } else {
        const int r = idx - SN;
        const int s = r >> 15;            // / E (E = 2^15)
        const int e = r & (E_ - 1);
        const int i = eidx[e];
        const int j = eidx[E_ + e];
        const float v = M[((size_t)s * N_ + i) * N_ + j];
        out[(size_t)N_ * FEAT_ + (size_t)e * FEAT_ + s * STEPS_ + t] = v;
    }
}

// ---------------------------------------------------------------------------
extern "C" void kernel_launch(void* const* d_in, const int* in_sizes, int n_in,
                              void* d_out, int out_size, void* d_ws, size_t ws_size,
                              hipStream_t stream)
{
    const float* coords     = (const float*)d_in[0];
    const float* log_sigmas = (const float*)d_in[1];
    const int*   eidx       = (const int*)d_in[2];
    float*       out        = (float*)d_out;

    const size_t SNN = (size_t)S_ * N_ * N_;   // 4,194,304 floats (16 MB)
    float* P    = (float*)d_ws;                // P (power 1)
    float* buf0 = P + SNN;                     // ping
    float* buf1 = P + 2 * SNN;                 // pong

    // Build transition matrices.
    build_p_kernel<<<S_ * N_, 256, 0, stream>>>(coords, log_sigmas, P);

    const int exTotal  = S_ * N_ + S_ * E_;
    const int exBlocks = (exTotal + 255) / 256;

    // k = 1: features from P itself (diag is exactly zero by construction).
    extract_kernel<<<exBlocks, 256, 0, stream>>>(P, eidx, out, 0);

    const dim3 grid(N_ / TN, N_ / TM, S_);
    const float* src = P;
    for (int t = 1; t < STEPS_; ++t) {
        float* dst = (t & 1) ? buf0 : buf1;
        gemm_step_kernel<<<grid, 256, 0, stream>>>(src, P, dst);
        extract_kernel<<<exBlocks, 256, 0, stream>>>(dst, eidx, out, t);
        src = dst;
    }
}